// IGCN_81312320847909
// MI455X (gfx1250) — compile-verified
//
#include <hip/hip_runtime.h>
#include <hip/hip_bf16.h>
#include <stdint.h>

// ---------------------------------------------------------------------------
// LightGCN graph diffusion on MI455X (gfx1250).
// Bandwidth/atomic-bound (AI ~ 0.25 FLOP/B) -> no WMMA; use TDM async tensor
// loads to stage contiguous edge streams into LDS, wave32 16-lane x float4
// row gathers, and hardware global f32 atomic adds for the scatter.
// ---------------------------------------------------------------------------

typedef unsigned int u32;
typedef unsigned int u32x4 __attribute__((ext_vector_type(4)));
typedef int          i32x8 __attribute__((ext_vector_type(8)));
typedef int          i32x4 __attribute__((ext_vector_type(4)));

#define TILE 2048              // edges staged per block
#define KEEP_INV (1.0f / 0.95f)

#if defined(__HIP_DEVICE_COMPILE__) && __has_builtin(__builtin_amdgcn_tensor_load_to_lds)
#define HAVE_TDM 1
#else
#define HAVE_TDM 0
#endif

__device__ __forceinline__ u32 lds_byte_off(const void* p) {
  // Generic pointers to LDS carry the LDS byte offset in the low 32 bits
  // (flat shared-aperture mapping).
  return (u32)(uintptr_t)p;
}

#if HAVE_TDM
// 1-D contiguous copy of `tile_elems` 4-byte elements from global -> LDS.
// tensor_dim0 = elements remaining so OOB tail reads return zero.
__device__ __forceinline__ void tdm_load_1d(u32 lds_off, const void* gptr,
                                            u32 remain_elems, u32 tile_elems) {
  unsigned long long ga = (unsigned long long)(uintptr_t)gptr;
  u32x4 g0;
  g0.x = 1u;                                   // count=1, user mode, no gather
  g0.y = lds_off;                              // lds_addr (bytes)
  g0.z = (u32)(ga & 0xFFFFFFFFu);              // global_addr[31:0]
  g0.w = (u32)((ga >> 32) & 0x1FFFFFFu)        // global_addr[56:32]
       | (2u << 30);                           // type = 2 ("image")
  const u32 dim0 = remain_elems;               // tensor_dim0 (elements)
  const u32 dim1 = 1u;                         // tensor_dim1
  i32x8 g1;
  g1.s0 = (int)(2u << 16);                     // data_size=2 (4B); wg_mask=0
  g1.s1 = (int)((dim0 & 0xFFFFu) << 16);       // [63:48] = tensor_dim0[15:0]
  g1.s2 = (int)(((dim0 >> 16) & 0xFFFFu)       // [79:64] = tensor_dim0[31:16]
              | ((dim1 & 0xFFFFu) << 16));     // [95:80] = tensor_dim1[15:0]
  g1.s3 = (int)(((dim1 >> 16) & 0xFFFFu)       // [111:96] = tensor_dim1[31:16]
              | ((tile_elems & 0xFFFFu) << 16)); // [127:112] = tile_dim0
  g1.s4 = 1;                                   // tile_dim1 = 1, tile_dim2 = 0
  g1.s5 = (int)tile_elems;                     // tensor_dim0_stride[31:0]
  g1.s6 = 0;                                   // stride hi / dim1_stride lo
  g1.s7 = 0;
  i32x4 z4; z4.x = 0; z4.y = 0; z4.z = 0; z4.w = 0;  // groups 2/3 unused (<=2D)
  i32x8 z8; z8.s0 = 0; z8.s1 = 0; z8.s2 = 0; z8.s3 = 0;
            z8.s4 = 0; z8.s5 = 0; z8.s6 = 0; z8.s7 = 0;
  __builtin_amdgcn_tensor_load_to_lds(g0, g1, z4, z4, z8, 0);
}
#endif

__device__ __forceinline__ void gadd_f32(float* p, float v) {
  // Relaxed agent-scope fetch_add with unused return -> no-return
  // global_atomic_add_f32 (STOREcnt path).
  (void)__hip_atomic_fetch_add(p, v, __ATOMIC_RELAXED, __HIP_MEMORY_SCOPE_AGENT);
}

// ---------------------------------------------------------------------------
// SpMM: y[dst] += (mask ? val/keep : 0) * x[src], y pre-zeroed.
// One block per TILE edges. Edge streams staged to LDS via TDM.
// 16 lanes x float4 per 64-dim row -> 2 edges per wave32.
// ---------------------------------------------------------------------------
__global__ __launch_bounds__(256) void k_spmm(
    const float* __restrict__ x, float* __restrict__ y,
    const int* __restrict__ esrc, const int* __restrict__ edst,
    const float* __restrict__ eval, const int* __restrict__ emsk, int E) {
  // NOT volatile: volatile blocks address-space inference and forces flat
  // loads. The asm memory clobber + barrier below keep the TDM-written data
  // observable, and plain addrspace(3) accesses lower to ds_load_b32.
  __shared__ int   s_src[TILE];
  __shared__ int   s_dst[TILE];
  __shared__ float s_val[TILE];
  __shared__ int   s_msk[TILE];

  const int tile0  = blockIdx.x * TILE;
  const u32 remain = (u32)(E - tile0);

#if HAVE_TDM
  if (threadIdx.x == 0) {   // one TDM issue per block (EXEC ignored by TDM)
    tdm_load_1d(lds_byte_off(s_src), esrc + tile0, remain, TILE);
    tdm_load_1d(lds_byte_off(s_dst), edst + tile0, remain, TILE);
    tdm_load_1d(lds_byte_off(s_val), eval + tile0, remain, TILE);
    tdm_load_1d(lds_byte_off(s_msk), emsk + tile0, remain, TILE);
    __builtin_amdgcn_s_wait_tensorcnt(0);
  }
  __asm__ volatile("" ::: "memory");   // LDS contents changed by TDM
  __syncthreads();
#else
  for (int i = threadIdx.x; i < TILE; i += 256) {
    const int e = tile0 + i;
    const bool ok = (u32)i < remain;
    s_src[i] = ok ? esrc[e] : 0;
    s_dst[i] = ok ? edst[e] : 0;
    s_val[i] = ok ? eval[e] : 0.0f;
    s_msk[i] = ok ? emsk[e] : 0;
  }
  __syncthreads();
#endif
  __asm__ volatile("" ::: "memory");   // don't fold/reorder the LDS reads

  const int sub   = threadIdx.x & 15;  // which float4 of the 64-dim row
  const int eslot = threadIdx.x >> 4;  // 16 edges in flight per pass

  for (int i = eslot; i < TILE; i += 16) {
    if ((u32)i < remain) {
      const int   s = s_src[i];      // ds_load_b32 (16-lane broadcast)
      const int   d = s_dst[i];
      const int   m = s_msk[i];
      float       v = s_val[i];
      v = m ? v * KEEP_INV : 0.0f;
      if (v != 0.0f) {
        const float4 t = ((const float4*)(x + (size_t)s * 64))[sub];  // b128
        float* o = y + (size_t)d * 64 + (size_t)sub * 4;
        gadd_f32(o + 0, t.x * v);
        gadd_f32(o + 1, t.y * v);
        gadd_f32(o + 2, t.z * v);
        gadd_f32(o + 3, t.w * v);
      }
    }
  }
}

// ---------------------------------------------------------------------------
// Elementwise helpers (float4, grid-stride).
// ---------------------------------------------------------------------------
__global__ void k_zero(float4* __restrict__ p, size_t n4) {
  const float4 z = make_float4(0.f, 0.f, 0.f, 0.f);
  for (size_t i = blockIdx.x * (size_t)blockDim.x + threadIdx.x; i < n4;
       i += (size_t)gridDim.x * blockDim.x)
    p[i] = z;
}

__global__ void k_copy(float4* __restrict__ o, const float4* __restrict__ a,
                       size_t n4) {
  for (size_t i = blockIdx.x * (size_t)blockDim.x + threadIdx.x; i < n4;
       i += (size_t)gridDim.x * blockDim.x)
    o[i] = a[i];
}

__global__ void k_acc_scale(float4* __restrict__ o, const float4* __restrict__ a,
                            size_t n4, float s) {
  for (size_t i = blockIdx.x * (size_t)blockDim.x + threadIdx.x; i < n4;
       i += (size_t)gridDim.x * blockDim.x) {
    float4 u = o[i];
    const float4 v = a[i];
    u.x = (u.x + v.x) * s; u.y = (u.y + v.y) * s;
    u.z = (u.z + v.z) * s; u.w = (u.w + v.w) * s;
    o[i] = u;
  }
}

// ---------------------------------------------------------------------------
// out = (emb + A emb + A^2 emb + A^3 emb) / 4, A = dropout-masked COO graph.
// ws layout: [ bufA : N*64 f32 | bufB : N*64 f32 ]  (512 MB)
// ---------------------------------------------------------------------------
extern "C" void kernel_launch(void* const* d_in, const int* in_sizes, int n_in,
                              void* d_out, int out_size, void* d_ws, size_t ws_size,
                              hipStream_t stream) {
  const float* emb  = (const float*)d_in[0];
  const int*   esrc = (const int*)d_in[1];
  const int*   edst = (const int*)d_in[2];
  const float* eval = (const float*)d_in[3];
  const int*   emsk = (const int*)d_in[4];

  const int    E  = in_sizes[1];
  const size_t NE = (size_t)in_sizes[0];   // N * 64 floats
  const size_t n4 = NE / 4;

  float* out = (float*)d_out;
  float* A   = (float*)d_ws;
  float* B   = A + NE;

  const int ntiles = (E + TILE - 1) / TILE;
  const dim3 eb(256), eg(4096);

  // out = emb (accumulator, scaled at the end)
  k_copy<<<eg, eb, 0, stream>>>((float4*)out, (const float4*)emb, n4);

  // layer 1: A = G * emb ; out += A
  k_zero<<<eg, eb, 0, stream>>>((float4*)A, n4);
  k_spmm<<<ntiles, 256, 0, stream>>>(emb, A, esrc, edst, eval, emsk, E);
  k_acc_scale<<<eg, eb, 0, stream>>>((float4*)out, (const float4*)A, n4, 1.0f);

  // layer 2: B = G * A ; out += B
  k_zero<<<eg, eb, 0, stream>>>((float4*)B, n4);
  k_spmm<<<ntiles, 256, 0, stream>>>(A, B, esrc, edst, eval, emsk, E);
  k_acc_scale<<<eg, eb, 0, stream>>>((float4*)out, (const float4*)B, n4, 1.0f);

  // layer 3: A = G * B ; out = (out + A) / 4
  k_zero<<<eg, eb, 0, stream>>>((float4*)A, n4);
  k_spmm<<<ntiles, 256, 0, stream>>>(B, A, esrc, edst, eval, emsk, E);
  k_acc_scale<<<eg, eb, 0, stream>>>((float4*)out, (const float4*)A, n4, 0.25f);
}